// TritterMLP_19653770346748
// MI455X (gfx1250) — compile-verified
//
#include <hip/hip_runtime.h>

typedef __attribute__((ext_vector_type(2))) float v2f;
typedef __attribute__((ext_vector_type(8))) float v8f;
typedef __attribute__((ext_vector_type(4))) unsigned v4u;
typedef __attribute__((ext_vector_type(4))) int v4i;
typedef __attribute__((ext_vector_type(8))) int v8i;

#define BM 128
#define BN 128
#define BK 16
#define LDA 20    // LDS stride (words) for A tile rows: 16 data + 4 pad (TDM pad)
#define LDB 136   // LDS stride (words) for B tile rows: 128 data + 8 pad (TDM pad)

// ---------------------------------------------------------------------------
// TDM: issue an async 2D tile load (global -> LDS) via the Tensor Data Mover.
//   - 4-byte elements (data_size=2)
//   - tile rows of tile_d0 elements, tile_d1 rows
//   - LDS row padding: pad_int_code -> 2^(code+1) DWORDs interval,
//     pad_amt_code -> (code+1) DWORDs of pad
// D# packing per CDNA5 ISA ch.10.8 (groups 2/3 zero: 2D tile).
// This toolchain's builtin is the 6-arg form:
//   (uint32x4 g0, int32x8 g1, int32x4 g2, int32x4 g3, int32x8 gx, i32 cpol)
// ---------------------------------------------------------------------------
__device__ __forceinline__ void tdm_load_2d(
    const void* gptr, unsigned lds_byte_addr,
    unsigned tensor_d0, unsigned tensor_d1, unsigned long long d0_stride,
    unsigned tile_d0, unsigned tile_d1,
    unsigned pad_int_code, unsigned pad_amt_code)
{
  const unsigned long long ga = (unsigned long long)(uintptr_t)gptr;
  v4u g0;
  g0.x = 1u;                                            // count=1, user desc
  g0.y = lds_byte_addr;                                 // lds_addr
  g0.z = (unsigned)(ga & 0xffffffffu);                  // global_addr[31:0]
  g0.w = (unsigned)((ga >> 32) & 0x01ffffffu)           // global_addr[56:32]
         | 0x80000000u;                                 // type=2 ("image")

  unsigned w0 = 0x00020000u                             // data_size=2 (4B)
              | (1u << 20)                              // pad_enable
              | (pad_int_code << 22)
              | (pad_amt_code << 25);
  v8i g1;
  g1[0] = (int)w0;
  g1[1] = (int)((tensor_d0 & 0xffffu) << 16);           // tensor_dim0[15:0]
  g1[2] = (int)((tensor_d0 >> 16) | ((tensor_d1 & 0xffffu) << 16));
  g1[3] = (int)((tensor_d1 >> 16) | (tile_d0 << 16));   // tile_dim0
  g1[4] = (int)(tile_d1 & 0xffffu);                     // tile_dim1; tile_dim2=0
  g1[5] = (int)(d0_stride & 0xffffffffull);             // tensor_dim0_stride
  g1[6] = (int)((d0_stride >> 32) & 0xffffull);         // hi bits; dim1_stride=0
  g1[7] = 0;
  const v4i gz4 = {0, 0, 0, 0};
  const v8i gz8 = {0, 0, 0, 0, 0, 0, 0, 0};
  __builtin_amdgcn_tensor_load_to_lds(g0, g1, gz4, gz4, gz8, 0);
}

__device__ __forceinline__ unsigned lds_addr_of(const void* p) {
  return (unsigned)(uintptr_t)p;   // low 32 bits of flat addr == LDS byte addr
}

// ---------------------------------------------------------------------------
// Kernel 1: fused gate/up GEMM + squared-relu epilogue
//   h[m,n] = squared_relu(x@Wg + bg) * (x@Wu + bu)
// M=8192, K=2048 (H), N=8192 (I)
// ---------------------------------------------------------------------------
__global__ __launch_bounds__(256) void gateup_kernel(
    const float* __restrict__ X, const float* __restrict__ Wg,
    const float* __restrict__ Bg, const float* __restrict__ Wu,
    const float* __restrict__ Bu, float* __restrict__ Hbuf,
    int M, int K, int N)
{
  __shared__ float As[2][BM * LDA];
  __shared__ float Bgs[2][BK * LDB];
  __shared__ float Bus[2][BK * LDB];

  const int tid  = threadIdx.x;
  const int lane = tid & 31;
  const int wave = tid >> 5;
  const int lh   = lane & 15;
  const int hb   = lane >> 4;
  const int wm   = (wave >> 2) << 6;
  const int wn   = (wave & 3) << 5;
  const int m0   = blockIdx.y * BM;
  const int n0   = blockIdx.x * BN;
  const int NT   = K / BK;

  v8f accg[4][2];
  v8f accu[4][2];
  const v8f zero8 = {0.f, 0.f, 0.f, 0.f, 0.f, 0.f, 0.f, 0.f};
#pragma unroll
  for (int sm = 0; sm < 4; ++sm)
#pragma unroll
    for (int sn = 0; sn < 2; ++sn) { accg[sm][sn] = zero8; accu[sm][sn] = zero8; }

  // one TDM issue = 3 async tile loads (A, Bg, Bu) into buffer `buf`
  auto issue_tile = [&](int kt, int buf) {
    const int k0 = kt * BK;
    // A tile: BMx16 rows of X, row pad 16DW -> +4DW (code 3 / code 3)
    tdm_load_2d(X + (size_t)m0 * K + k0, lds_addr_of(&As[buf][0]),
                (unsigned)K, (unsigned)M, (unsigned long long)K,
                BK, BM, 3u, 3u);
    // B tiles: 16xBN rows of W, row pad 128DW -> +8DW (code 6 / code 7)
    tdm_load_2d(Wg + (size_t)k0 * N + n0, lds_addr_of(&Bgs[buf][0]),
                (unsigned)N, (unsigned)K, (unsigned long long)N,
                BN, BK, 6u, 7u);
    tdm_load_2d(Wu + (size_t)k0 * N + n0, lds_addr_of(&Bus[buf][0]),
                (unsigned)N, (unsigned)K, (unsigned long long)N,
                BN, BK, 6u, 7u);
  };

  if (wave == 0) issue_tile(0, 0);

  for (int kt = 0; kt < NT; ++kt) {
    const int buf = kt & 1;
    if (wave == 0) {
      if (kt + 1 < NT) {
        issue_tile(kt + 1, buf ^ 1);                   // prefetch next tile
        __builtin_amdgcn_s_wait_tensorcnt(3);          // tile kt's 3 ops done
      } else {
        __builtin_amdgcn_s_wait_tensorcnt(0);
      }
    }
    __syncthreads();

    const float* __restrict__ Asb = As[buf];
    const float* __restrict__ Bgb = Bgs[buf];
    const float* __restrict__ Bub = Bus[buf];

#pragma unroll
    for (int ks = 0; ks < 4; ++ks) {
      const int k = ks * 4 + 2 * hb;
      v2f afrag[4];
#pragma unroll
      for (int sm = 0; sm < 4; ++sm)
        afrag[sm] = *(const v2f*)(Asb + (wm + sm * 16 + lh) * LDA + k);
#pragma unroll
      for (int sn = 0; sn < 2; ++sn) {
        const int n = wn + sn * 16 + lh;
        v2f bgf, buf2;
        bgf.x  = Bgb[k * LDB + n];
        bgf.y  = Bgb[(k + 1) * LDB + n];
        buf2.x = Bub[k * LDB + n];
        buf2.y = Bub[(k + 1) * LDB + n];
#pragma unroll
        for (int sm = 0; sm < 4; ++sm) {
          accg[sm][sn] = __builtin_amdgcn_wmma_f32_16x16x4_f32(
              false, afrag[sm], false, bgf, (short)0, accg[sm][sn], false, false);
          accu[sm][sn] = __builtin_amdgcn_wmma_f32_16x16x4_f32(
              false, afrag[sm], false, buf2, (short)0, accu[sm][sn], false, false);
        }
      }
    }
    __syncthreads();
  }

  // epilogue: bias, squared-relu, product
#pragma unroll
  for (int sm = 0; sm < 4; ++sm) {
#pragma unroll
    for (int sn = 0; sn < 2; ++sn) {
      const int col = n0 + wn + sn * 16 + lh;
      const float bgv = Bg[col];
      const float buv = Bu[col];
#pragma unroll
      for (int g = 0; g < 8; ++g) {
        const int row = m0 + wm + sm * 16 + hb * 8 + g;
        float gate = accg[sm][sn][g] + bgv;
        float up   = accu[sm][sn][g] + buv;
        gate = gate * fmaxf(gate, 0.0f);   // x * relu(x)
        Hbuf[(size_t)row * N + col] = gate * up;
      }
    }
  }
}

// ---------------------------------------------------------------------------
// Kernel 2: down GEMM   out = h @ Wd + bd
// M=8192, K=8192 (I), N=2048 (H)
// ---------------------------------------------------------------------------
__global__ __launch_bounds__(256) void down_kernel(
    const float* __restrict__ Hbuf, const float* __restrict__ Wd,
    const float* __restrict__ Bd, float* __restrict__ Out,
    int M, int K, int N)
{
  __shared__ float As[2][BM * LDA];
  __shared__ float Bs[2][BK * LDB];

  const int tid  = threadIdx.x;
  const int lane = tid & 31;
  const int wave = tid >> 5;
  const int lh   = lane & 15;
  const int hb   = lane >> 4;
  const int wm   = (wave >> 2) << 6;
  const int wn   = (wave & 3) << 5;
  const int m0   = blockIdx.y * BM;
  const int n0   = blockIdx.x * BN;
  const int NT   = K / BK;

  v8f acc[4][2];
  const v8f zero8 = {0.f, 0.f, 0.f, 0.f, 0.f, 0.f, 0.f, 0.f};
#pragma unroll
  for (int sm = 0; sm < 4; ++sm)
#pragma unroll
    for (int sn = 0; sn < 2; ++sn) acc[sm][sn] = zero8;

  auto issue_tile = [&](int kt, int buf) {
    const int k0 = kt * BK;
    tdm_load_2d(Hbuf + (size_t)m0 * K + k0, lds_addr_of(&As[buf][0]),
                (unsigned)K, (unsigned)M, (unsigned long long)K,
                BK, BM, 3u, 3u);
    tdm_load_2d(Wd + (size_t)k0 * N + n0, lds_addr_of(&Bs[buf][0]),
                (unsigned)N, (unsigned)K, (unsigned long long)N,
                BN, BK, 6u, 7u);
  };

  if (wave == 0) issue_tile(0, 0);

  for (int kt = 0; kt < NT; ++kt) {
    const int buf = kt & 1;
    if (wave == 0) {
      if (kt + 1 < NT) {
        issue_tile(kt + 1, buf ^ 1);
        __builtin_amdgcn_s_wait_tensorcnt(2);          // tile kt's 2 ops done
      } else {
        __builtin_amdgcn_s_wait_tensorcnt(0);
      }
    }
    __syncthreads();

    const float* __restrict__ Asb = As[buf];
    const float* __restrict__ Bsb = Bs[buf];

#pragma unroll
    for (int ks = 0; ks < 4; ++ks) {
      const int k = ks * 4 + 2 * hb;
      v2f afrag[4];
#pragma unroll
      for (int sm = 0; sm < 4; ++sm)
        afrag[sm] = *(const v2f*)(Asb + (wm + sm * 16 + lh) * LDA + k);
#pragma unroll
      for (int sn = 0; sn < 2; ++sn) {
        const int n = wn + sn * 16 + lh;
        v2f bf;
        bf.x = Bsb[k * LDB + n];
        bf.y = Bsb[(k + 1) * LDB + n];
#pragma unroll
        for (int sm = 0; sm < 4; ++sm) {
          acc[sm][sn] = __builtin_amdgcn_wmma_f32_16x16x4_f32(
              false, afrag[sm], false, bf, (short)0, acc[sm][sn], false, false);
        }
      }
    }
    __syncthreads();
  }

#pragma unroll
  for (int sm = 0; sm < 4; ++sm) {
#pragma unroll
    for (int sn = 0; sn < 2; ++sn) {
      const int col = n0 + wn + sn * 16 + lh;
      const float bdv = Bd[col];
#pragma unroll
      for (int g = 0; g < 8; ++g) {
        const int row = m0 + wm + sm * 16 + hb * 8 + g;
        Out[(size_t)row * N + col] = acc[sm][sn][g] + bdv;
      }
    }
  }
}

// ---------------------------------------------------------------------------
extern "C" void kernel_launch(void* const* d_in, const int* in_sizes, int n_in,
                              void* d_out, int out_size, void* d_ws, size_t ws_size,
                              hipStream_t stream) {
  const float* X  = (const float*)d_in[0];  // (B,S,H) = (4,2048,2048)
  const float* Wg = (const float*)d_in[1];  // (H,I)
  const float* bg = (const float*)d_in[2];  // (I)
  const float* Wu = (const float*)d_in[3];  // (H,I)
  const float* bu = (const float*)d_in[4];  // (I)
  const float* Wd = (const float*)d_in[5];  // (I,H)
  const float* bd = (const float*)d_in[6];  // (H)
  float* out  = (float*)d_out;              // (B,S,H)
  float* hbuf = (float*)d_ws;               // (M,I) fp32 = 256 MB scratch

  const int M = 4 * 2048;   // B*S tokens
  const int H = 2048;
  const int I = 8192;

  dim3 blk(256, 1, 1);
  dim3 g1(I / BN, M / BM, 1);  // 64 x 64
  gateup_kernel<<<g1, blk, 0, stream>>>(X, Wg, bg, Wu, bu, hbuf, M, H, I);

  dim3 g2(H / BN, M / BM, 1);  // 16 x 64
  down_kernel<<<g2, blk, 0, stream>>>(hbuf, Wd, bd, out, M, I, H);
}